// KeypointMatchingLoss_55035710931707
// MI455X (gfx1250) — compile-verified
//
#include <hip/hip_runtime.h>
#include <math.h>

// ---------------------------------------------------------------------------
// KeypointMatchingLoss for MI455X (gfx1250, wave32).
// Streaming-reduction workload (~263 MB @ 23.3 TB/s => ~11us floor).
// Uses the CDNA5 async-to-LDS path (ASYNCcnt) to double-buffer each point's
// 768B KNN tile while the wave computes distances / argmin / logsumexp.
// ---------------------------------------------------------------------------

#define WPB 8            // waves per block (256 threads)
#define NB_DEFAULT 2048  // blocks for the main kernel

typedef int v4i __attribute__((__vector_size__(16)));
typedef int v2i __attribute__((__vector_size__(8)));
typedef __attribute__((address_space(1))) v4i gv4i;
typedef __attribute__((address_space(3))) v4i lv4i;
typedef __attribute__((address_space(1))) v2i gv2i;
typedef __attribute__((address_space(3))) v2i lv2i;

__device__ __forceinline__ void async_copy_b128(const void* g, void* l) {
#if __has_builtin(__builtin_amdgcn_global_load_async_to_lds_b128)
  __builtin_amdgcn_global_load_async_to_lds_b128((gv4i*)g, (lv4i*)l, 0, 0);
#else
  unsigned lofs = (unsigned)(unsigned long long)(__attribute__((address_space(3))) void*)l;
  asm volatile("global_load_async_to_lds_b128 %0, %1, off" :: "v"(lofs), "v"(g) : "memory");
#endif
}

__device__ __forceinline__ void async_copy_b64(const void* g, void* l) {
#if __has_builtin(__builtin_amdgcn_global_load_async_to_lds_b64)
  __builtin_amdgcn_global_load_async_to_lds_b64((gv2i*)g, (lv2i*)l, 0, 0);
#else
  unsigned lofs = (unsigned)(unsigned long long)(__attribute__((address_space(3))) void*)l;
  asm volatile("global_load_async_to_lds_b64 %0, %1, off" :: "v"(lofs), "v"(g) : "memory");
#endif
}

__device__ __forceinline__ void wait_async_le2() {
#if __has_builtin(__builtin_amdgcn_s_wait_asynccnt)
  __builtin_amdgcn_s_wait_asynccnt(2);
#else
  asm volatile("s_wait_asynccnt 0x2" ::: "memory");
#endif
}

__device__ __forceinline__ void wait_async_le0() {
#if __has_builtin(__builtin_amdgcn_s_wait_asynccnt)
  __builtin_amdgcn_s_wait_asynccnt(0);
#else
  asm volatile("s_wait_asynccnt 0x0" ::: "memory");
#endif
}

__device__ __forceinline__ void wait_ds0() {
#if __has_builtin(__builtin_amdgcn_s_wait_dscnt)
  __builtin_amdgcn_s_wait_dscnt(0);
#else
  asm volatile("s_wait_dscnt 0x0" ::: "memory");
#endif
}

// Stage one point's 64x3 f32 KNN tile (768B) into LDS: 32 lanes x (16B + 8B).
__device__ __forceinline__ void issue_tile(const float* __restrict__ knn,
                                           long long p, float* lbuf, int lane) {
  const char* g = (const char*)(knn + (size_t)p * 192);
  char* l = (char*)lbuf;
  async_copy_b128(g + 16 * lane, l + 16 * lane);          // bytes [0,512)
  async_copy_b64(g + 512 + 8 * lane, l + 512 + 8 * lane); // bytes [512,768)
}

__global__ __launch_bounds__(256) void kml_main(
    const float* __restrict__ corres, const float* __restrict__ knn,
    const float* __restrict__ mlog, const float* __restrict__ conf,
    const float* __restrict__ gt, float* __restrict__ partials,
    long long nPts, int ptsPerWave) {
  __shared__ __align__(16) float stage[WPB][2][192]; // 2 x 768B per wave
  __shared__ float red[WPB][4];

  const int tid = threadIdx.x;
  const int wv = tid >> 5;
  const int lane = tid & 31;

  const long long wgid = (long long)blockIdx.x * WPB + wv;
  const long long first = wgid * (long long)ptsPerWave;
  int mc = 0;
  if (first < nPts) {
    long long r = nPts - first;
    mc = (r < (long long)ptsPerWave) ? (int)r : ptsPerWave;
  }

  float accCnt = 0.f, accFeat = 0.f, accCorr = 0.f, accOv = 0.f;

  if (mc > 0) issue_tile(knn, first, &stage[wv][0][0], lane);

  for (int j = 0; j < mc; ++j) {
    const long long p = first + j;
    if (j + 1 < mc) {
      wait_ds0(); // prior iteration's LDS reads retired before reuse
      issue_tile(knn, p + 1, &stage[wv][(j + 1) & 1][0], lane);
      wait_async_le2(); // current tile's 2 async ops complete (in-order)
    } else {
      wait_async_le0();
    }
    const float* q = &stage[wv][j & 1][0];

    // wave-uniform per-point data
    const float* c = corres + (size_t)p * 6;
    const float tx = c[0], ty = c[1], tz = c[2];
    const float X = c[3], Y = c[4], Z = c[5];
    const float sx = gt[0] * X + gt[1] * Y + gt[2] * Z + gt[3];
    const float sy = gt[4] * X + gt[5] * Y + gt[6] * Z + gt[7];
    const float sz = gt[8] * X + gt[9] * Y + gt[10] * Z + gt[11];

    // lane's two neighbors: k = 2*lane, 2*lane+1
    const float ax = q[6 * lane + 0], ay = q[6 * lane + 1], az = q[6 * lane + 2];
    const float bx = q[6 * lane + 3], by = q[6 * lane + 4], bz = q[6 * lane + 5];
    const float da = sqrtf((sx - ax) * (sx - ax) + (sy - ay) * (sy - ay) + (sz - az) * (sz - az));
    const float db = sqrtf((sx - bx) * (sx - bx) + (sy - by) * (sy - by) + (sz - bz) * (sz - bz));

    const float2* l2 = (const float2*)(mlog + (size_t)p * 64);
    const float2 ml = l2[lane];

    // wave argmin over (dist, k) with first-index tie-break
    float dm = da;
    int km = 2 * lane;
    if (db < dm) { dm = db; km = 2 * lane + 1; }
    for (int o = 16; o > 0; o >>= 1) {
      float od = __shfl_xor(dm, o);
      int ok = __shfl_xor(km, o);
      if (od < dm || (od == dm && ok < km)) { dm = od; km = ok; }
    }
    const bool maskp = dm < 0.2f;

    const bool iga = (da < 0.4f) && (2 * lane != km);
    const bool igb = (db < 0.4f) && (2 * lane + 1 != km);
    const float la = ml.x - (iga ? 10000.f : 0.f);
    const float lb = ml.y - (igb ? 10000.f : 0.f);

    float lm = fmaxf(la, lb);
    for (int o = 16; o > 0; o >>= 1) lm = fmaxf(lm, __shfl_xor(lm, o));
    float e = expf(la - lm) + expf(lb - lm);
    for (int o = 16; o > 0; o >>= 1) e += __shfl_xor(e, o);
    const float lse = lm + logf(e);
    const float pl = __shfl((km & 1) ? ml.y : ml.x, km >> 1); // logits[idx1] (never ignored)

    const float dc = sqrtf((sx - tx) * (sx - tx) + (sy - ty) * (sy - ty) + (sz - tz) * (sz - tz));
    const float pp = conf[p];
    const float lgp = fmaxf(logf(pp), -100.f);
    const float l1p = fmaxf(log1pf(-pp), -100.f);
    const bool lab = dc < 0.2f;
    const bool wgt = lab || !(dc < 0.4f);
    const float bce = lab ? -lgp : -l1p;

    if (maskp) { accCnt += 1.f; accFeat += (lse - pl); accCorr += dc; }
    if (wgt) accOv += bce;
  }

  if (lane == 0) {
    red[wv][0] = accCnt; red[wv][1] = accFeat;
    red[wv][2] = accCorr; red[wv][3] = accOv;
  }
  __syncthreads();
  if (tid < 4) { // fixed-order, deterministic block reduction
    float s = 0.f;
    for (int w = 0; w < WPB; ++w) s += red[w][tid];
    partials[(size_t)blockIdx.x * 4 + tid] = s;
  }
}

__global__ __launch_bounds__(256) void kml_finalize(const float* __restrict__ partials,
                                                    int nb, float* __restrict__ out,
                                                    float fN) {
  __shared__ double sd[256 * 4];
  const int t = threadIdx.x;
  double a0 = 0, a1 = 0, a2 = 0, a3 = 0;
  for (int b = t; b < nb; b += 256) {
    a0 += (double)partials[4 * b + 0];
    a1 += (double)partials[4 * b + 1];
    a2 += (double)partials[4 * b + 2];
    a3 += (double)partials[4 * b + 3];
  }
  sd[t] = a0; sd[256 + t] = a1; sd[512 + t] = a2; sd[768 + t] = a3;
  __syncthreads();
  for (int s = 128; s > 0; s >>= 1) {
    if (t < s) {
      sd[t] += sd[t + s];
      sd[256 + t] += sd[256 + t + s];
      sd[512 + t] += sd[512 + t + s];
      sd[768 + t] += sd[768 + t + s];
    }
    __syncthreads();
  }
  if (t == 0) {
    const float cnt = (float)sd[0];
    const float feat = (float)sd[256];
    const float corr = (float)sd[512];
    const float ov = (float)sd[768];
    const float denom = fmaxf(cnt, 1.f);
    out[0] = (cnt > 0.f) ? feat / denom : 0.f; // loss_feat
    out[1] = ov / fN;                          // loss_ov (mean over N)
    out[2] = (cnt > 0.f) ? corr / denom : 0.f; // loss_corr
  }
}

extern "C" void kernel_launch(void* const* d_in, const int* in_sizes, int n_in,
                              void* d_out, int out_size, void* d_ws, size_t ws_size,
                              hipStream_t stream) {
  (void)n_in; (void)out_size;
  const float* corres = (const float*)d_in[0];
  const float* knn    = (const float*)d_in[1];
  const float* mlog   = (const float*)d_in[2];
  const float* conf   = (const float*)d_in[3];
  const float* gt     = (const float*)d_in[4];

  const long long N = (long long)(in_sizes[0] / 6);

  int nb = NB_DEFAULT;
  if (ws_size < (size_t)nb * 4 * sizeof(float)) {
    nb = (int)(ws_size / (4 * sizeof(float)));
    if (nb < 1) nb = 1;
  }
  const long long totalWaves = (long long)nb * WPB;
  const int ppw = (int)((N + totalWaves - 1) / totalWaves);

  kml_main<<<nb, 256, 0, stream>>>(corres, knn, mlog, conf, gt,
                                   (float*)d_ws, N, ppw);
  kml_finalize<<<1, 256, 0, stream>>>((const float*)d_ws, nb, (float*)d_out, (float)N);
}